// HungarianMatcher_82987358093468
// MI455X (gfx1250) — compile-verified
//
#include <hip/hip_runtime.h>
#include <hip/hip_bf16.h>
#include <cfloat>

// Problem constants (from reference setup_inputs)
#define BS 64
#define NQ 900
#define NT 200
#define NC 92            // multiple of 4 -> clean K-loop for WMMA f32 16x16x4
#define KSTEPS (NC / 4)  // 23 WMMA steps per 16x16 tile
#define TTILES ((NT + 15) / 16)  // 13
#define QTILES ((NQ + 15) / 16)  // 57

typedef float v2f __attribute__((ext_vector_type(2)));
typedef float v8f __attribute__((ext_vector_type(8)));

// ---------------------------------------------------------------------------
// Kernel 1: row softmax of pred_logits (BS*NQ rows of NC=92) -> prob in d_ws.
// One wave32 per row; each lane covers elements {lane, lane+32, lane+64}.
// ---------------------------------------------------------------------------
__global__ __launch_bounds__(256) void softmax_rows_k(
    const float* __restrict__ logits, float* __restrict__ prob) {
  const int row = blockIdx.x * 8 + (threadIdx.x >> 5);
  if (row >= BS * NQ) return;
  const int lane = threadIdx.x & 31;
  const float* x = logits + (size_t)row * NC;

  float a0 = x[lane];
  float a1 = x[lane + 32];
  float a2 = (lane + 64 < NC) ? x[lane + 64] : -FLT_MAX;

  float m = fmaxf(a0, fmaxf(a1, a2));
#pragma unroll
  for (int off = 16; off > 0; off >>= 1) m = fmaxf(m, __shfl_xor(m, off, 32));

  float e0 = __expf(a0 - m);
  float e1 = __expf(a1 - m);
  float e2 = (lane + 64 < NC) ? __expf(a2 - m) : 0.0f;

  float s = e0 + e1 + e2;
#pragma unroll
  for (int off = 16; off > 0; off >>= 1) s += __shfl_xor(s, off, 32);

  const float inv = 1.0f / s;
  float* p = prob + (size_t)row * NC;
  p[lane]      = e0 * inv;
  p[lane + 32] = e1 * inv;
  if (lane + 64 < NC) p[lane + 64] = e2 * inv;
}

// ---------------------------------------------------------------------------
// Kernel 2: strip kernel. One wave owns a full 16 x NT output strip for one
// q-tile. A fragments (16xNC prob tile) and the lane's 8 pred_boxes are loaded
// ONCE into registers, then reused across all 13 t-tiles. Per t-tile the inner
// loop is pure VALU (one-hot synth) + 23 chained V_WMMA_F32_16X16X4_F32.
//
// A layout (16x4 f32): lane m = lane&15; VGPR0/1 hold K = 2*(lane>>4) + {0,1}.
// B layout (4x16 f32): lane n = lane&15; VGPR0/1 hold K = 2*(lane>>4) + {0,1}.
// C/D layout (16x16 f32): element v of v8f -> (m = v + 8*(lane>>4), n = lane&15).
// ---------------------------------------------------------------------------
__global__ __launch_bounds__(128) void cost_strip_k(
    const float* __restrict__ prob,
    const float* __restrict__ pred_boxes,
    const float* __restrict__ boxes,
    const int*   __restrict__ labels,
    float* __restrict__ out) {
  const int wave = threadIdx.x >> 5;
  const int qt = blockIdx.x * 4 + wave;
  if (qt * 16 >= NQ) return;          // wave-uniform exit; EXEC stays full otherwise
  const int b = blockIdx.z;

  const int lane = threadIdx.x & 31;
  const int half = lane >> 4;         // 0: lanes 0-15, 1: lanes 16-31
  const int l15  = lane & 15;
  const int kb   = half << 1;         // K sub-offset {0,2} per A/B f32 layout

  // ---- load the full A strip for this wave's q-tile, once -------------------
  const int q_m  = qt * 16 + l15;
  const int q_cl = (q_m < NQ) ? q_m : (NQ - 1);
  const float* arow = prob + ((size_t)b * NQ + q_cl) * NC;

  v2f afrag[KSTEPS];
#pragma unroll
  for (int kk = 0; kk < KSTEPS; ++kk) {
    afrag[kk].x = arow[4 * kk + kb];
    afrag[kk].y = arow[4 * kk + kb + 1];
  }

  // ---- load this lane's 8 pred boxes (rows of the C tile it owns), once ----
  float4 pbs[8];
#pragma unroll
  for (int v = 0; v < 8; ++v) {
    const int m = v + (half << 3);
    const int q = qt * 16 + m;
    const int qc = (q < NQ) ? q : (NQ - 1);
    pbs[v] = *(const float4*)(pred_boxes + ((size_t)b * NQ + qc) * 4);
  }

  // ---- sweep the 13 t-tiles, reusing A + pred boxes from registers ----------
  for (int tt = 0; tt < TTILES; ++tt) {
    const int t_n   = tt * 16 + l15;
    const int t_cl  = (t_n < NT) ? t_n : (NT - 1);
    const int label = labels[(size_t)b * NT + t_cl];

    v8f acc = {0.f, 0.f, 0.f, 0.f, 0.f, 0.f, 0.f, 0.f};
#pragma unroll
    for (int kk = 0; kk < KSTEPS; ++kk) {
      const int k = 4 * kk;
      v2f bm;                         // one-hot column of labels: exact select
      bm.x = (label == (k + kb))     ? 1.0f : 0.0f;
      bm.y = (label == (k + kb + 1)) ? 1.0f : 0.0f;
      acc = __builtin_amdgcn_wmma_f32_16x16x4_f32(
          /*neg_a=*/false, afrag[kk], /*neg_b=*/false, bm,
          /*c_mod=*/(short)0, acc, /*reuse_a=*/false, /*reuse_b=*/false);
    }

    // ---- epilogue: per-element bbox L1 + GIoU, combine, store ---------------
    const float4 tb = *(const float4*)(boxes + ((size_t)b * NT + t_cl) * 4);
    const float bx0 = tb.x - 0.5f * tb.z, by0 = tb.y - 0.5f * tb.w;
    const float bx1 = tb.x + 0.5f * tb.z, by1 = tb.y + 0.5f * tb.w;
    const float area2 = (bx1 - bx0) * (by1 - by0);
    const bool  tval  = (t_n < NT);

#pragma unroll
    for (int v = 0; v < 8; ++v) {
      const int m = v + (half << 3);
      const int q = qt * 16 + m;
      if (q >= NQ || !tval) continue;
      const float4 pb = pbs[v];

      const float l1 = fabsf(pb.x - tb.x) + fabsf(pb.y - tb.y) +
                       fabsf(pb.z - tb.z) + fabsf(pb.w - tb.w);

      const float ax0 = pb.x - 0.5f * pb.z, ay0 = pb.y - 0.5f * pb.w;
      const float ax1 = pb.x + 0.5f * pb.z, ay1 = pb.y + 0.5f * pb.w;
      const float area1 = (ax1 - ax0) * (ay1 - ay0);

      const float iw    = fminf(ax1, bx1) - fmaxf(ax0, bx0);
      const float ih    = fminf(ay1, by1) - fmaxf(ay0, by0);
      const float inter = fmaxf(iw, 0.f) * fmaxf(ih, 0.f);
      const float uni   = area1 + area2 - inter;
      const float iou   = inter / uni;

      const float ew    = fmaxf(ax1, bx1) - fminf(ax0, bx0);
      const float eh    = fmaxf(ay1, by1) - fminf(ay0, by0);
      const float areae = fmaxf(ew, 0.f) * fmaxf(eh, 0.f);
      const float giou  = iou - (areae - uni) / areae;

      // COST_BBOX*l1 + COST_CLASS*(-prob_gathered) + COST_GIOU*(-giou)
      out[((size_t)b * NQ + q) * NT + t_n] = 5.0f * l1 - acc[v] - 2.0f * giou;
    }
  }
}

// ---------------------------------------------------------------------------
extern "C" void kernel_launch(void* const* d_in, const int* in_sizes, int n_in,
                              void* d_out, int out_size, void* d_ws, size_t ws_size,
                              hipStream_t stream) {
  const float* pred_logits = (const float*)d_in[0];  // (64, 900, 92)
  const float* pred_boxes  = (const float*)d_in[1];  // (64, 900, 4)
  const float* boxes       = (const float*)d_in[2];  // (64, 200, 4)
  const int*   labels      = (const int*)d_in[3];    // (64, 200)
  float* out  = (float*)d_out;                       // (64, 900, 200)
  float* prob = (float*)d_ws;                        // BS*NQ*NC floats (~20.2 MB)

  // 1) softmax rows -> prob
  const int rows = BS * NQ;
  softmax_rows_k<<<(rows + 7) / 8, 256, 0, stream>>>(pred_logits, prob);

  // 2) fused WMMA class-cost + bbox/giou epilogue; one wave per 16xNT strip
  dim3 grid((QTILES + 3) / 4, 1, BS);
  cost_strip_k<<<grid, 128, 0, stream>>>(prob, pred_boxes, boxes, labels, out);
}